// HybridGNN_7576322310634
// MI455X (gfx1250) — compile-verified
//
#include <hip/hip_runtime.h>
#include <math.h>

#define U_NODES 25000
#define I_NODES 25000
#define NN      50000          // total nodes
#define EMB     64
#define XDIM    128            // 2*EMB
#define CDIM    300
#define NE      800000
#define NET     850000         // NE + NN self loops
#define NB      25000

typedef float v2f __attribute__((ext_vector_type(2)));
typedef float v8f __attribute__((ext_vector_type(8)));

// ---------------------------------------------------------------- utilities

__global__ void fill_kernel(float* __restrict__ p, float v, int n) {
  int i = blockIdx.x * blockDim.x + threadIdx.x;
  if (i < n) p[i] = v;
}

__device__ __forceinline__ void atomic_max_float(float* addr, float v) {
  // sign-split trick: lowers to global_atomic_max_i32 / global_atomic_min_u32
  if (v >= 0.0f) atomicMax((int*)addr, __float_as_int(v));
  else           atomicMin((unsigned int*)addr, __float_as_uint(v));
}

// -------------------------------------------------- generic f32 WMMA GEMM
// C[M,NCOLS] = act(A[M,KK] @ B[KK,NCOLS] + bias); one 16x16 tile per wave32,
// K step 4 via v_wmma_f32_16x16x4_f32 (fp32 parity with the reference).
// All strides/shapes are compile-time so the K loop reduces to constant-offset
// global_load_b64 + pointer increments feeding back-to-back WMMAs.
template <int LDA, int LDB, int LDC, int NCOLS, int KK, int ACT, bool HASBIAS>
__global__ __launch_bounds__(256)
void wmma_gemm_kernel(const float* __restrict__ A, const float* __restrict__ B,
                      const float* __restrict__ bias, float* __restrict__ C,
                      int M) {
  constexpr int TILES_N = NCOLS / 16;
  int wave = (int)((blockIdx.x * blockDim.x + threadIdx.x) >> 5);
  int lane = threadIdx.x & 31;
  int tiles_m = (M + 15) >> 4;
  if (wave >= tiles_m * TILES_N) return;   // wave-uniform exit: EXEC stays all-1
  int tm = wave / TILES_N, tn = wave % TILES_N;

  int half = lane >> 4;                    // 0 => lanes 0-15, 1 => lanes 16-31
  int l15  = lane & 15;
  int arow = tm * 16 + l15; if (arow >= M) arow = M - 1;   // clamp keeps EXEC full
  int col  = tn * 16 + l15;

  // ISA f32 A 16x4 layout: VGPR0={K0 | K2}, VGPR1={K1 | K3} across lane halves.
  const float* ap = A + (size_t)arow * LDA + half * 2;
  const float* bp = B + (size_t)(half * 2) * LDB + col;

  v8f acc = {};
#pragma unroll 4
  for (int k = 0; k < KK; k += 4) {
    v2f a; a.x = ap[0];   a.y = ap[1];       // contiguous -> global_load_b64
    v2f b; b.x = bp[0];   b.y = bp[LDB];     // two rows of B, constant offset
    ap += 4;
    bp += 4 * LDB;
    __builtin_prefetch(ap + 32, 0, 1);
    acc = __builtin_amdgcn_wmma_f32_16x16x4_f32(false, a, false, b,
                                                (short)0, acc, false, false);
  }

  float bv = HASBIAS ? bias[col] : 0.0f;
#pragma unroll
  for (int r = 0; r < 8; ++r) {
    int row = tm * 16 + r + half * 8;        // C/D layout: lanes16-31 hold M=r+8
    if (row < M) {
      float v = acc[r] + bv;
      if (ACT == 1) v = fmaxf(v, 0.0f);      // relu
      C[(size_t)row * LDC + col] = v;
    }
  }
}

// -------------------------------------------------------------- node setup
// users: [user_emb | 0], items: [item_emb | (content proj, filled by GEMM)]
__global__ void build_x_kernel(const float* __restrict__ ue,
                               const float* __restrict__ ie,
                               float* __restrict__ x) {
  long i = (long)blockIdx.x * blockDim.x + threadIdx.x;
  if (i >= (long)NN * XDIM) return;
  int n = (int)(i / XDIM), c = (int)(i % XDIM);
  if (n < U_NODES)      x[i] = (c < EMB) ? ue[(size_t)n * EMB + c] : 0.0f;
  else if (c < EMB)     x[i] = ie[(size_t)(n - U_NODES) * EMB + c];
  // item right half written later by the content-projection GEMM
}

// ------------------------------------------------------------- attention
__global__ void alpha_kernel(const float* __restrict__ hm, int ldh, int H,
                             const float* __restrict__ a_src,
                             const float* __restrict__ a_dst,
                             float* __restrict__ als, float* __restrict__ ald) {
  int t = blockIdx.x * blockDim.x + threadIdx.x;   // over NN*H
  if (t >= NN * H) return;
  int n = t / H, hh = t % H;
  const float* hp = hm + (size_t)n * ldh + hh * EMB;
  const float* as = a_src + hh * EMB;
  const float* ad = a_dst + hh * EMB;
  float s = 0.0f, d = 0.0f;
  for (int k = 0; k < EMB; ++k) { float hv = hp[k]; s += hv * as[k]; d += hv * ad[k]; }
  als[t] = s; ald[t] = d;
}

// pass 1: e = leaky_relu(als[src]+ald[dst]); segment max via atomics
__global__ void edge_pass1_kernel(const int* __restrict__ ei, int H,
                                  const float* __restrict__ als,
                                  const float* __restrict__ ald,
                                  float* __restrict__ ebuf,
                                  float* __restrict__ mbuf) {
  int t = blockIdx.x * blockDim.x + threadIdx.x;   // over NET*H
  if (t >= NET * H) return;
  int e = t / H, hh = t % H;
  int src, dst;
  if (e < NE) { src = ei[e]; dst = ei[NE + e]; }
  else        { src = dst = e - NE; }              // self loop
  float v = als[src * H + hh] + ald[dst * H + hh];
  v = v > 0.0f ? v : 0.2f * v;                     // leaky relu, slope 0.2
  ebuf[t] = v;
  atomic_max_float(&mbuf[dst * H + hh], v);
}

// pass 2: ex = exp(e - m[dst]); segment sum
__global__ void edge_pass2_kernel(const int* __restrict__ ei, int H,
                                  const float* __restrict__ mbuf,
                                  float* __restrict__ ebuf,
                                  float* __restrict__ den) {
  int t = blockIdx.x * blockDim.x + threadIdx.x;
  if (t >= NET * H) return;
  int e = t / H, hh = t % H;
  int dst = (e < NE) ? ei[NE + e] : e - NE;
  float ex = expf(ebuf[t] - mbuf[dst * H + hh]);
  ebuf[t] = ex;
  atomicAdd(&den[dst * H + hh], ex);
}

// pass 3: agg[dst] += coef * h[src]  (one wave32 per (edge, head), 2 floats/lane)
__global__ void edge_pass3_kernel(const int* __restrict__ ei, int H, int ldh,
                                  const float* __restrict__ hm,
                                  const float* __restrict__ ebuf,
                                  const float* __restrict__ den,
                                  float* __restrict__ agg) {
  int w = (int)((blockIdx.x * blockDim.x + threadIdx.x) >> 5);
  int lane = threadIdx.x & 31;
  if (w >= NET * H) return;
  int e = w / H, hh = w % H;
  int src, dst;
  if (e < NE) { src = ei[e]; dst = ei[NE + e]; }
  else        { src = dst = e - NE; }
  float coef = ebuf[w] / (den[dst * H + hh] + 1e-16f);
  const float* hp = hm  + (size_t)src * ldh + hh * EMB;
  float*       op = agg + (size_t)dst * ldh + hh * EMB;
  atomicAdd(&op[lane],      coef * hp[lane]);
  atomicAdd(&op[lane + 32], coef * hp[lane + 32]);
}

// conv1 epilogue: x2 = elu(agg + b1)
__global__ void finalize1_kernel(const float* __restrict__ agg,
                                 const float* __restrict__ b1,
                                 float* __restrict__ x) {
  long i = (long)blockIdx.x * blockDim.x + threadIdx.x;
  if (i >= (long)NN * XDIM) return;
  float v = agg[i] + b1[(int)(i % XDIM)];
  x[i] = v > 0.0f ? v : (expf(v) - 1.0f);
}

// conv2 epilogue: x3 = agg + b2 (no activation)
__global__ void finalize2_kernel(const float* __restrict__ agg,
                                 const float* __restrict__ b2,
                                 float* __restrict__ x) {
  long i = (long)blockIdx.x * blockDim.x + threadIdx.x;
  if (i >= (long)NN * EMB) return;
  x[i] = agg[i] + b2[(int)(i % EMB)];
}

// combined = [u_emb | item_gnn | user_gnn]
__global__ void build_comb_kernel(const int* __restrict__ uid,
                                  const int* __restrict__ iid,
                                  const float* __restrict__ ue,
                                  const float* __restrict__ x3,
                                  float* __restrict__ comb) {
  long i = (long)blockIdx.x * blockDim.x + threadIdx.x;
  if (i >= (long)NB * 192) return;
  int b = (int)(i / 192), c = (int)(i % 192);
  float v;
  if (c < 64)        v = ue[(size_t)uid[b] * 64 + c];
  else if (c < 128)  v = x3[(size_t)(U_NODES + iid[b]) * 64 + (c - 64)];
  else               v = x3[(size_t)uid[b] * 64 + (c - 128)];
  comb[i] = v;
}

// out[b] = hid[b,:] . p2_w + p2_b
__global__ void predict_out_kernel(const float* __restrict__ hid,
                                   const float* __restrict__ p2w,
                                   const float* __restrict__ p2b,
                                   float* __restrict__ out) {
  int b = blockIdx.x * blockDim.x + threadIdx.x;
  if (b >= NB) return;
  float acc = p2b[0];
  for (int j = 0; j < 64; ++j) acc += hid[(size_t)b * 64 + j] * p2w[j];
  out[b] = acc;
}

// ---------------------------------------------------------------- launcher

extern "C" void kernel_launch(void* const* d_in, const int* in_sizes, int n_in,
                              void* d_out, int out_size, void* d_ws, size_t ws_size,
                              hipStream_t stream) {
  (void)in_sizes; (void)n_in; (void)out_size; (void)ws_size;
  const int*   uid = (const int*)d_in[0];
  const int*   iid = (const int*)d_in[1];
  const float* cf  = (const float*)d_in[2];
  const int*   ei  = (const int*)d_in[3];
  const float* uew = (const float*)d_in[4];
  const float* iew = (const float*)d_in[5];
  const float* cpw = (const float*)d_in[6];
  const float* cpb = (const float*)d_in[7];
  const float* W1  = (const float*)d_in[8];
  const float* a1s = (const float*)d_in[9];
  const float* a1d = (const float*)d_in[10];
  const float* b1  = (const float*)d_in[11];
  const float* W2  = (const float*)d_in[12];
  const float* a2s = (const float*)d_in[13];
  const float* a2d = (const float*)d_in[14];
  const float* b2  = (const float*)d_in[15];
  const float* p1w = (const float*)d_in[16];
  const float* p1b = (const float*)d_in[17];
  const float* p2w = (const float*)d_in[18];
  const float* p2b = (const float*)d_in[19];
  float* out = (float*)d_out;

  float* ws = (float*)d_ws;
  size_t o = 0;
  float* X   = ws + o; o += (size_t)NN * XDIM;   // x / x2 / x3
  float* Hm  = ws + o; o += (size_t)NN * XDIM;   // h (conv1), h2 (conv2 reuses)
  float* AGG = ws + o; o += (size_t)NN * XDIM;   // aggregation target
  float* ALS = ws + o; o += (size_t)NN * 2;
  float* ALD = ws + o; o += (size_t)NN * 2;
  float* MB  = ws + o; o += (size_t)NN * 2;
  float* DEN = ws + o; o += (size_t)NN * 2;
  float* EB  = ws + o; o += (size_t)NET * 2;
  float* CMB = ws + o; o += (size_t)NB * 192;
  float* HID = ws + o; o += (size_t)NB * 64;

  const int T = 256;
  auto cdiv = [](long a, long b) { return (int)((a + b - 1) / b); };

  // ---- node features ----
  build_x_kernel<<<cdiv((long)NN * XDIM, T), T, 0, stream>>>(uew, iew, X);
  {
    int tiles = cdiv(I_NODES, 16) * (EMB / 16);
    // content proj: A=[25000,300] lda=300, B=[300,64] ldb=64 -> item right half
    wmma_gemm_kernel<CDIM, EMB, XDIM, EMB, CDIM, 0, true>
        <<<cdiv((long)tiles * 32, T), T, 0, stream>>>(
            cf, cpw, cpb, X + (size_t)U_NODES * XDIM + EMB, I_NODES);
  }

  // ---- conv1 (heads=2) ----
  {
    int tiles = cdiv(NN, 16) * (XDIM / 16);
    wmma_gemm_kernel<XDIM, XDIM, XDIM, XDIM, XDIM, 0, false>
        <<<cdiv((long)tiles * 32, T), T, 0, stream>>>(X, W1, nullptr, Hm, NN);
  }
  alpha_kernel<<<cdiv((long)NN * 2, T), T, 0, stream>>>(Hm, XDIM, 2, a1s, a1d, ALS, ALD);
  fill_kernel<<<cdiv((long)NN * 2, T), T, 0, stream>>>(MB, -INFINITY, NN * 2);
  fill_kernel<<<cdiv((long)NN * 2, T), T, 0, stream>>>(DEN, 0.0f, NN * 2);
  fill_kernel<<<cdiv((long)NN * XDIM, T), T, 0, stream>>>(AGG, 0.0f, NN * XDIM);
  edge_pass1_kernel<<<cdiv((long)NET * 2, T), T, 0, stream>>>(ei, 2, ALS, ALD, EB, MB);
  edge_pass2_kernel<<<cdiv((long)NET * 2, T), T, 0, stream>>>(ei, 2, MB, EB, DEN);
  edge_pass3_kernel<<<cdiv((long)NET * 2 * 32, T), T, 0, stream>>>(ei, 2, XDIM, Hm, EB, DEN, AGG);
  finalize1_kernel<<<cdiv((long)NN * XDIM, T), T, 0, stream>>>(AGG, b1, X);

  // ---- conv2 (heads=1) ----
  {
    int tiles = cdiv(NN, 16) * (EMB / 16);
    wmma_gemm_kernel<XDIM, EMB, EMB, EMB, XDIM, 0, false>
        <<<cdiv((long)tiles * 32, T), T, 0, stream>>>(X, W2, nullptr, Hm, NN);
  }
  alpha_kernel<<<cdiv((long)NN, T), T, 0, stream>>>(Hm, EMB, 1, a2s, a2d, ALS, ALD);
  fill_kernel<<<cdiv((long)NN, T), T, 0, stream>>>(MB, -INFINITY, NN);
  fill_kernel<<<cdiv((long)NN, T), T, 0, stream>>>(DEN, 0.0f, NN);
  fill_kernel<<<cdiv((long)NN * EMB, T), T, 0, stream>>>(AGG, 0.0f, NN * EMB);
  edge_pass1_kernel<<<cdiv((long)NET, T), T, 0, stream>>>(ei, 1, ALS, ALD, EB, MB);
  edge_pass2_kernel<<<cdiv((long)NET, T), T, 0, stream>>>(ei, 1, MB, EB, DEN);
  edge_pass3_kernel<<<cdiv((long)NET * 32, T), T, 0, stream>>>(ei, 1, EMB, Hm, EB, DEN, AGG);
  finalize2_kernel<<<cdiv((long)NN * EMB, T), T, 0, stream>>>(AGG, b2, X);

  // ---- prediction MLP ----
  build_comb_kernel<<<cdiv((long)NB * 192, T), T, 0, stream>>>(uid, iid, uew, X, CMB);
  {
    int tiles = cdiv(NB, 16) * (EMB / 16);
    wmma_gemm_kernel<192, EMB, EMB, EMB, 192, 1, true>
        <<<cdiv((long)tiles * 32, T), T, 0, stream>>>(CMB, p1w, p1b, HID, NB);
  }
  predict_out_kernel<<<cdiv((long)NB, T), T, 0, stream>>>(HID, p2w, p2b, out);
}